// PhysicalDiffraction_55843164782805
// MI455X (gfx1250) — compile-verified
//
#include <hip/hip_runtime.h>
#include <math.h>

// CDNA5 / gfx1250: wave32, WMMA 16x16x4 F32.
typedef __attribute__((ext_vector_type(2))) float v2f;
typedef __attribute__((ext_vector_type(8))) float v8f;

#define FFT_N     512
#define NN        (FFT_N * FFT_N)       // 262144 elements per plane
#define NIMG      128                   // B*C = 8*16
#define CHUNK     16                    // 2 cplx ping-pong bufs = 67MB -> L2 resident
#define BLK_M     128                   // 8 waves x 16 rows
#define TILE_N2   32                    // per-wave N extent (2 x 16 accum sets)
#define KT        32                    // K panel depth
#define KSTEPS    (KT / 4)              // 8 WMMA k-steps per panel
#define NPANEL    (FFT_N / KT)          // 16 panels
#define KSTRIDE   (KT + 4)              // 36 dwords: banks 4*(9n mod 16) -> conflict-free b64

#define WMMA_F32(a, b, c) \
    __builtin_amdgcn_wmma_f32_16x16x4_f32(false, (a), false, (b), (short)0, (c), false, false)

// ---------------------------------------------------------------------------
// Forward DFT matrix F = exp(-2*pi*i*j*k/N) and inverse Fi = conj(F)/N.
// ---------------------------------------------------------------------------
__global__ __launch_bounds__(256) void init_dft(float* __restrict__ fre,
                                                float* __restrict__ fim,
                                                float* __restrict__ ire,
                                                float* __restrict__ iim) {
    int idx = blockIdx.x * 256 + threadIdx.x;       // 0..NN-1
    int j = idx >> 9;
    int k = idx & (FFT_N - 1);
    int m = (j * k) & (FFT_N - 1);
    float ang = -6.2831853071795864769f * (float)m * (1.0f / (float)FFT_N);
    float s, c;
    __sincosf(ang, &s, &c);
    fre[idx] = c;
    fim[idx] = s;
    ire[idx] = c * (1.0f / (float)FFT_N);
    iim[idx] = -s * (1.0f / (float)FFT_N);
}

// ---------------------------------------------------------------------------
// One transform stage: D = W (512x512 complex DFT matrix) x S (512x512),
// output written TRANSPOSED. Block = 256 thr = 8 waves; block tile 128x32,
// each wave a 16x32 tile (two 16x16 accumulator sets sharing A fragments).
// A fragments stream from the L2-resident W matrix. The 32x32 complex B
// panel is double-buffered in LDS, stored K-transposed so each WMMA B
// fragment is a single aligned ds_load_b64 (bank-conflict-free, no repack
// moves). Next panel's global loads are register-staged across the whole
// WMMA panel for latency hiding; one barrier per panel.
// ---------------------------------------------------------------------------
template <bool REAL_IN, bool APPLY_H, bool REAL_OUT>
__global__ __launch_bounds__(256) void dft_stage(
    const float* __restrict__ Wre, const float* __restrict__ Wim,
    const float* __restrict__ Sre, const float* __restrict__ Sim,
    const float* __restrict__ Hre, const float* __restrict__ Him,
    float* __restrict__ Dre, float* __restrict__ Dim,
    long sInStride, long sOutStride) {
    __shared__ float ldsRe[2][TILE_N2][KSTRIDE];   // [buf][col][k]
    __shared__ float ldsIm[2][TILE_N2][KSTRIDE];

    const int tid   = threadIdx.x;
    const int lane  = tid & 31;
    const int wave  = tid >> 5;
    const int lhalf = lane >> 4;        // 0: lanes 0-15, 1: lanes 16-31
    const int lmod  = lane & 15;

    const int img = blockIdx.z;
    const int n0  = blockIdx.x * TILE_N2;            // output column base
    const int m0  = blockIdx.y * BLK_M + wave * 16;  // output row base (wave)

    const float* sre = Sre + (long)img * sInStride;
    const float* sim = Sim + (long)img * sInStride;

    // 16x4 f32 A layout: lanes 0-15 -> K = k..k+1, lanes 16-31 -> K = k+2..k+3,
    // row M = lane % 16.
    const int arow = m0 + lmod;
    const float* wreRow = Wre + (long)arow * FFT_N + 2 * lhalf;
    const float* wimRow = Wim + (long)arow * FFT_N + 2 * lhalf;

    // Per-thread staging: element e = tid + q*256 -> (kk = e>>5, nn = e&31).
    const int stKK[4] = { tid >> 5, (tid + 256) >> 5, (tid + 512) >> 5, (tid + 768) >> 5 };
    const int stNN    = tid & 31;

    float stRe[4], stIm[4];

    // Issue global loads (+ fused Hc filter) for panel starting at k0 into regs.
    auto issueLoads = [&](int k0) {
#pragma unroll
        for (int q = 0; q < 4; ++q) {
            long gidx = (long)(k0 + stKK[q]) * FFT_N + (n0 + stNN);
            float xre = sre[gidx];
            float xim = REAL_IN ? 0.0f : sim[gidx];
            if (APPLY_H) {
                float hr = Hre[gidx];
                float hi = Him[gidx];
                float tre = xre * hr - xim * hi;
                xim       = xre * hi + xim * hr;
                xre       = tre;
            }
            stRe[q] = xre;
            if (!REAL_IN) stIm[q] = xim;
        }
    };
    // Commit staged registers into LDS buffer (K-transposed layout).
    auto writeLds = [&](int buf) {
#pragma unroll
        for (int q = 0; q < 4; ++q) {
            ldsRe[buf][stNN][stKK[q]] = stRe[q];
            if (!REAL_IN) ldsIm[buf][stNN][stKK[q]] = stIm[q];
        }
    };

    v8f rr0 = {}, ii0 = {}, ri0 = {}, ir0 = {};
    v8f rr1 = {}, ii1 = {}, ri1 = {}, ir1 = {};

    // Prologue: panel 0 into buffer 0.
    issueLoads(0);
    writeLds(0);
    __syncthreads();

    for (int p = 0; p < NPANEL; ++p) {
        const int k0 = p * KT;
        if (p + 1 < NPANEL) issueLoads(k0 + KT);   // next panel loads in flight

        // Batch all A fragments for this panel (16 b64 loads in flight).
        v2f areA[KSTEPS], aimA[KSTEPS];
#pragma unroll
        for (int i = 0; i < KSTEPS; ++i) {
            areA[i] = *(const v2f*)(wreRow + k0 + 4 * i);
            aimA[i] = *(const v2f*)(wimRow + k0 + 4 * i);
        }

        const int buf = p & 1;
#pragma unroll
        for (int i = 0; i < KSTEPS; ++i) {
            const int kb = 4 * i + 2 * lhalf;   // 4x16 B halves mirror A halves
            v2f bre0 = *(const v2f*)&ldsRe[buf][lmod][kb];
            v2f bre1 = *(const v2f*)&ldsRe[buf][lmod + 16][kb];
            rr0 = WMMA_F32(areA[i], bre0, rr0);
            ir0 = WMMA_F32(aimA[i], bre0, ir0);
            rr1 = WMMA_F32(areA[i], bre1, rr1);
            ir1 = WMMA_F32(aimA[i], bre1, ir1);
            if (!REAL_IN) {
                v2f bim0 = *(const v2f*)&ldsIm[buf][lmod][kb];
                v2f bim1 = *(const v2f*)&ldsIm[buf][lmod + 16][kb];
                ii0 = WMMA_F32(aimA[i], bim0, ii0);
                ri0 = WMMA_F32(areA[i], bim0, ri0);
                ii1 = WMMA_F32(aimA[i], bim1, ii1);
                ri1 = WMMA_F32(areA[i], bim1, ri1);
            }
        }

        if (p + 1 < NPANEL) writeLds((p + 1) & 1); // other buffer: no conflict
        __syncthreads();
    }

    // Complex combine: Dre = rr - ii, Dim = ri + ir (real input: ii = ri = 0).
    v8f dre0 = REAL_IN ? rr0 : (rr0 - ii0);
    v8f dim0 = REAL_IN ? ir0 : (ri0 + ir0);
    v8f dre1 = REAL_IN ? rr1 : (rr1 - ii1);
    v8f dim1 = REAL_IN ? ir1 : (ri1 + ir1);

    // C/D layout: VGPR g -> M = m0 + 8*lhalf + g, N = n0 + lane%16 (+16 tile1).
    // Transposed store out[N][M]: 2x float4 per lane per tile per plane.
    const int  mbase = m0 + 8 * lhalf;
    const long obase = (long)img * sOutStride + mbase;
    float* dr0 = Dre + obase + (long)(n0 + lmod) * FFT_N;
    float* dr1 = Dre + obase + (long)(n0 + lmod + 16) * FFT_N;
    *(float4*)(dr0)     = make_float4(dre0[0], dre0[1], dre0[2], dre0[3]);
    *(float4*)(dr0 + 4) = make_float4(dre0[4], dre0[5], dre0[6], dre0[7]);
    *(float4*)(dr1)     = make_float4(dre1[0], dre1[1], dre1[2], dre1[3]);
    *(float4*)(dr1 + 4) = make_float4(dre1[4], dre1[5], dre1[6], dre1[7]);
    if (!REAL_OUT) {
        float* di0 = Dim + obase + (long)(n0 + lmod) * FFT_N;
        float* di1 = Dim + obase + (long)(n0 + lmod + 16) * FFT_N;
        *(float4*)(di0)     = make_float4(dim0[0], dim0[1], dim0[2], dim0[3]);
        *(float4*)(di0 + 4) = make_float4(dim0[4], dim0[5], dim0[6], dim0[7]);
        *(float4*)(di1)     = make_float4(dim1[0], dim1[1], dim1[2], dim1[3]);
        *(float4*)(di1 + 4) = make_float4(dim1[4], dim1[5], dim1[6], dim1[7]);
    }
}

// ---------------------------------------------------------------------------
// Four staged transposed left-multiplies per image chunk:
//   S1: buf1 = (F  x)^T                    (real input)
//   S2: buf2 = (F  buf1)^T = fft2(x)
//   S3: buf1 = (Fi (buf2 .* Hc))^T
//   S4: y    = real( (Fi buf1)^T ) = real(ifft2(fft2(x) .* Hc))
// ---------------------------------------------------------------------------
extern "C" void kernel_launch(void* const* d_in, const int* in_sizes, int n_in,
                              void* d_out, int out_size, void* d_ws, size_t ws_size,
                              hipStream_t stream) {
    (void)in_sizes; (void)n_in; (void)out_size; (void)ws_size;

    const float* x   = (const float*)d_in[0];   // [8,16,512,512] f32
    const float* Hre = (const float*)d_in[1];   // [512,512] f32
    const float* Him = (const float*)d_in[2];   // [512,512] f32
    float*       out = (float*)d_out;           // [8,16,512,512] f32

    float* ws   = (float*)d_ws;
    float* Fre  = ws;
    float* Fim  = ws + 1L * NN;
    float* Ire  = ws + 2L * NN;
    float* Iim  = ws + 3L * NN;
    float* buf1 = ws + 4L * NN;                    // CHUNK * 2 planes
    float* buf2 = buf1 + (long)CHUNK * 2L * NN;    // CHUNK * 2 planes

    init_dft<<<NN / 256, 256, 0, stream>>>(Fre, Fim, Ire, Iim);

    const long planeStride = (long)NN;     // real tensor: 1 plane / image
    const long cplxStride  = 2L * NN;      // complex buffers: re+im planes
    dim3 grid(FFT_N / TILE_N2, FFT_N / BLK_M, CHUNK);  // (16, 4, 16)

    for (int c0 = 0; c0 < NIMG; c0 += CHUNK) {
        const float* xin  = x   + (long)c0 * NN;
        float*       yout = out + (long)c0 * NN;

        dft_stage<true, false, false><<<grid, 256, 0, stream>>>(
            Fre, Fim, xin, xin, Hre, Him, buf1, buf1 + NN,
            planeStride, cplxStride);

        dft_stage<false, false, false><<<grid, 256, 0, stream>>>(
            Fre, Fim, buf1, buf1 + NN, Hre, Him, buf2, buf2 + NN,
            cplxStride, cplxStride);

        dft_stage<false, true, false><<<grid, 256, 0, stream>>>(
            Ire, Iim, buf2, buf2 + NN, Hre, Him, buf1, buf1 + NN,
            cplxStride, cplxStride);

        dft_stage<false, false, true><<<grid, 256, 0, stream>>>(
            Ire, Iim, buf1, buf1 + NN, Hre, Him, yout, yout,
            cplxStride, planeStride);
    }
}